// QKProjection_13932873908329
// MI455X (gfx1250) — compile-verified
//
#include <hip/hip_runtime.h>

// QKProjection: O = Q + tril(K K^T) Q   (B=4, L=4096, D=1024, fp32)
// Fused causal flash-style kernel using V_WMMA_F32_16X16X4_F32,
// TDM (tensor_load_to_lds) staging of the K row-tile, ping-pong LDS
// score buffer (one barrier per chunk), NT stores for the output.

typedef __attribute__((ext_vector_type(2))) float v2f;
typedef __attribute__((ext_vector_type(8))) float v8f;
typedef __attribute__((ext_vector_type(4))) unsigned int v4u;
typedef __attribute__((ext_vector_type(8))) int v8i;
typedef __attribute__((ext_vector_type(4))) int v4i;

#define BATCH   4
#define SEQ     4096
#define DIM     1024
#define M_TILE  16      // rows of t per workgroup
#define S_CHUNK 128     // s columns processed per chunk (16 per wave)
#define NWAVES  8
#define NTHREADS (NWAVES * 32)

__global__ __launch_bounds__(NTHREADS)
void qkproj_kernel(const float* __restrict__ Q,
                   const float* __restrict__ K,
                   float* __restrict__ Out)
{
    __shared__ float sK[M_TILE * DIM];          // 64 KB: K row-tile, staged once
    __shared__ float sS[2][M_TILE * S_CHUNK];   // 16 KB: ping-pong score chunks

    const int b        = blockIdx.y;
    // Heavy (high row_base) tiles launch first -> better tail behavior.
    const int row_base = (gridDim.x - 1 - blockIdx.x) * M_TILE;
    const int tid  = threadIdx.x;
    const int lane = tid & 31;
    const int wave = tid >> 5;
    const int m    = lane & 15;   // M/N index within fragment
    const int h    = lane >> 4;   // k-half selector (ISA 16x4/4x16 layouts)

    const float* Kb = K + (size_t)b * SEQ * DIM;
    const float* Qb = Q + (size_t)b * SEQ * DIM;

    // ---- Stage the 16 K rows of this tile into LDS (reused every chunk) ----
#if __has_builtin(__builtin_amdgcn_tensor_load_to_lds) && \
    __has_builtin(__builtin_amdgcn_s_wait_tensorcnt)
    if (wave == 0) {
        // Tensor DMA: one 16384-element (64 KB) contiguous tile -> LDS.
        const unsigned long long ga =
            (unsigned long long)(uintptr_t)(Kb + (size_t)row_base * DIM);
        const unsigned lds_off = (unsigned)(uintptr_t)(void*)sK; // low 32b = LDS offset
        v4u g0;
        g0[0] = 1u;                                   // count=1, user descriptor
        g0[1] = lds_off;                              // lds_addr
        g0[2] = (unsigned)(ga & 0xFFFFFFFFu);         // global_addr[31:0]
        g0[3] = (unsigned)((ga >> 32) & 0x01FFFFFFu)  // global_addr[56:32]
                | 0x80000000u;                        // type=2 ("image")
        v8i g1;
        g1[0] = 0x00020000;                 // data_size=4B, no mask/pad/iterate
        g1[1] = (int)0x40000000u;           // tensor_dim0 = 16384 (low16 @ [63:48])
        g1[2] = (int)0x00010000u;           // tensor_dim0 hi=0, tensor_dim1 = 1
        g1[3] = (int)0x40000000u;           // tile_dim0 = 16384 @ [127:112]
        g1[4] = 1;                          // tile_dim1 = 1
        g1[5] = 16384;                      // tensor_dim0_stride = 16384
        g1[6] = 0;
        g1[7] = 0;
        v4i gz; gz[0] = 0; gz[1] = 0; gz[2] = 0; gz[3] = 0;
#if defined(__clang_major__) && (__clang_major__ >= 23)
        v8i g4; for (int i = 0; i < 8; ++i) g4[i] = 0;
        __builtin_amdgcn_tensor_load_to_lds(g0, g1, gz, gz, g4, 0);
#else
        __builtin_amdgcn_tensor_load_to_lds(g0, g1, gz, gz, 0);
#endif
        __builtin_amdgcn_s_wait_tensorcnt(0);
    }
#else
    {
        const float4* gsrc = (const float4*)(Kb + (size_t)row_base * DIM);
        float4* ldst = (float4*)sK;
        for (int i = tid; i < M_TILE * DIM / 4; i += NTHREADS)
            ldst[i] = gsrc[i];
    }
#endif
    __syncthreads();

    // Persistent output accumulators: wave owns d in [128*wave, 128*wave+128)
    v8f acc[8];
    #pragma unroll
    for (int j = 0; j < 8; ++j)
        #pragma unroll
        for (int v = 0; v < 8; ++v)
            acc[j][v] = 0.0f;

    const int row_end = row_base + M_TILE - 1;

    for (int s0 = 0; s0 <= row_end; s0 += S_CHUNK) {
        const int pb = (s0 >> 7) & 1;   // ping-pong buffer select

        // ---- Phase A: S_chunk(16x128) = Krow(16xD) * Kcol^T, fp32 WMMA ----
        // wave computes score columns [s0 + 16*wave, s0 + 16*wave + 16)
        const int scol = s0 + 16 * wave;
        v8f s_acc;
        #pragma unroll
        for (int v = 0; v < 8; ++v) s_acc[v] = 0.0f;

        const float* krow_lane = sK + m * DIM + 2 * h;                  // A frag (LDS)
        const float* kcol_lane = Kb + (size_t)(scol + m) * DIM + 2 * h; // B frag (L2)
        #pragma unroll 8
        for (int d0 = 0; d0 < DIM; d0 += 4) {
            v2f a  = *(const v2f*)(krow_lane + d0);  // A[m, d0+2h], A[m, d0+2h+1]
            v2f bb = *(const v2f*)(kcol_lane + d0);  // B[2h, n=m],  B[2h+1, n=m]
            s_acc = __builtin_amdgcn_wmma_f32_16x16x4_f32(
                        false, a, false, bb, (short)0, s_acc, false, false);
        }

        // Causal mask + spill to LDS.  C layout: lane n+16h, VGPR v -> (M=v+8h, N=n)
        #pragma unroll
        for (int v = 0; v < 8; ++v) {
            const int t = row_base + v + 8 * h;
            const int s = scol + m;
            sS[pb][(v + 8 * h) * S_CHUNK + 16 * wave + m] = (s <= t) ? s_acc[v] : 0.0f;
        }
        __syncthreads();   // single barrier per chunk (ping-pong makes it safe)

        // Prefetch next chunk's K columns into the cache hierarchy.
        if (s0 + S_CHUNK <= row_end)
            __builtin_prefetch(Kb + (size_t)(s0 + S_CHUNK + 16 * wave + m) * DIM + 64 * h, 0, 1);

        // ---- Phase B: acc += S_chunk(16x128) * Q[s-chunk, wave's 128 d] ----
        const int d_base = 128 * wave;
        const float* sSp = sS[pb];
        #pragma unroll 2
        for (int ks = 0; ks < S_CHUNK; ks += 4) {
            v2f a;
            a.x = sSp[m * S_CHUNK + ks + 2 * h];      // A[m, ks+2h]
            a.y = sSp[m * S_CHUNK + ks + 2 * h + 1];  // A[m, ks+2h+1]
            const float* q0 = Qb + (size_t)(s0 + ks + 2 * h) * DIM + d_base + m;
            #pragma unroll
            for (int j = 0; j < 8; ++j) {
                v2f bb;
                bb.x = q0[16 * j];        // B[2h,   n] = Q[s0+ks+2h,   d_base+16j+n]
                bb.y = q0[16 * j + DIM];  // B[2h+1, n] = Q[s0+ks+2h+1, d_base+16j+n]
                acc[j] = __builtin_amdgcn_wmma_f32_16x16x4_f32(
                             false, a, false, bb, (short)0, acc[j], false, false);
            }
        }
    }

    // ---- Epilogue: Out = Q + acc  (NT stores: output is never re-read) ----
    {
        const int d_base = 128 * wave;
        float* Ob = Out + (size_t)b * SEQ * DIM;
        #pragma unroll
        for (int j = 0; j < 8; ++j) {
            #pragma unroll
            for (int v = 0; v < 8; ++v) {
                const int t = row_base + v + 8 * h;
                const int d = d_base + 16 * j + m;
                const size_t idx = (size_t)t * DIM + d;
                __builtin_nontemporal_store(Qb[idx] + acc[j][v], &Ob[idx]);
            }
        }
    }
}

extern "C" void kernel_launch(void* const* d_in, const int* in_sizes, int n_in,
                              void* d_out, int out_size, void* d_ws, size_t ws_size,
                              hipStream_t stream) {
    const float* Q = (const float*)d_in[0];
    const float* K = (const float*)d_in[1];
    float* Out = (float*)d_out;
    (void)in_sizes; (void)n_in; (void)out_size; (void)d_ws; (void)ws_size;

    dim3 grid(SEQ / M_TILE, BATCH);   // 256 row-tiles x 4 batches
    dim3 block(NTHREADS);             // 8 waves of 32
    qkproj_kernel<<<grid, block, 0, stream>>>(Q, K, Out);
}